// TaskAlignedAssigner_45526653337972
// MI455X (gfx1250) — compile-verified
//
#include <hip/hip_runtime.h>
#include <math.h>

#define BS   8
#define NA   33600
#define M_GT 64
#define NC   80
#define TOPK 13
#define TILES 2100            // NA / 16
#define EPSF 1e-9f
#define PI_F 3.14159265358979323846f

typedef float v2f __attribute__((ext_vector_type(2)));
typedef float v8f __attribute__((ext_vector_type(8)));

// ---------------- workspace layout (bytes) ----------------
#define AM_BYTES   ((size_t)BS * M_GT * NA * 4)   // 68,812,800 (masked align metric)
#define MASK_BYTES ((size_t)BS * NA * 8)          //  2,150,400
#define POS_BYTES  ((size_t)BS * M_GT * 4)        //      2,048
#define BA_BYTES   ((size_t)BS * NA * 4)          //  1,075,200
#define WS_AMB  ((size_t)0)
#define WS_MASK (WS_AMB  + AM_BYTES)
#define WS_PA   (WS_MASK + MASK_BYTES)
#define WS_PO   (WS_PA   + POS_BYTES)
#define WS_AMS  (WS_PO   + POS_BYTES)
#define WS_LAB  (WS_AMS  + BA_BYTES)
#define WS_TGT  (WS_LAB  + BA_BYTES)

// ---------------- output layout (floats, concat in return order) ------------
#define OFF_LAB 0
#define OFF_CIR (BS * NA)                         //    268,800
#define OFF_SCO (OFF_CIR + BS * NA * 3)          //  1,075,200
#define OFF_FG  (OFF_SCO + BS * NA * NC)         // 22,579,200
#define OFF_IDX (OFF_FG + BS * NA)               // 22,848,000

// ============================================================================
// Branch-free math helpers (straight-line VALU; no exec divergence)
// ============================================================================
// Abramowitz & Stegun 4.4.46: |err| <= ~2e-8 rad on [-1, 1]
__device__ __forceinline__ float fast_acosf(float x) {
    const float xa = fabsf(x);
    float p = fmaf(xa, -0.0012624911f, 0.0066700901f);
    p = fmaf(xa, p, -0.0170881256f);
    p = fmaf(xa, p,  0.0308918810f);
    p = fmaf(xa, p, -0.0501743046f);
    p = fmaf(xa, p,  0.0889789874f);
    p = fmaf(xa, p, -0.2145988016f);
    p = fmaf(xa, p,  1.5707963050f);
    const float r = __builtin_amdgcn_sqrtf(fmaxf(1.0f - xa, 0.0f)) * p;
    return (x >= 0.0f) ? r : (PI_F - r);
}

// circle IoU given raw squared center distance (may be tiny-negative from WMMA)
__device__ __forceinline__ float circle_iou(float d2raw, float r0, float r1) {
    const float r0sq = r0 * r0, r1sq = r1 * r1;
    const float d    = __builtin_amdgcn_sqrtf(fmaxf(d2raw, 0.0f) + 1e-12f);
    const float d1v  = (r0sq - r1sq + d * d) * __builtin_amdgcn_rcpf(2.0f * d);
    const float d2v  = d - d1v;
    float c0 = d1v * __builtin_amdgcn_rcpf(fmaxf(r0, EPSF));
    c0 = fminf(fmaxf(c0, -1.0f), 1.0f);
    float c1 = d2v * __builtin_amdgcn_rcpf(fmaxf(r1, EPSF));
    c1 = fminf(fmaxf(c1, -1.0f), 1.0f);
    const float a0v = r0sq * fast_acosf(c0)
                    - d1v * __builtin_amdgcn_sqrtf(fmaxf(r0sq - d1v * d1v, 0.0f));
    const float a1v = r1sq * fast_acosf(c1)
                    - d2v * __builtin_amdgcn_sqrtf(fmaxf(r1sq - d2v * d2v, 0.0f));
    const float rmin = fminf(r0, r1);
    float inter = (d <= fabsf(r0 - r1)) ? (PI_F * rmin * rmin) : (a0v + a1v);
    inter = (d >= r0 + r1) ? 0.0f : inter;
    const float uni = PI_F * (r0sq + r1sq) - inter;
    return (uni > 0.0f) ? inter * __builtin_amdgcn_rcpf(uni) : 0.0f;
}

// ============================================================================
// K0: zero bitmask + pos accumulators (contiguous region)
// ============================================================================
__global__ void k0_zero(unsigned int* __restrict__ p, int nwords) {
    int i = blockIdx.x * blockDim.x + threadIdx.x;
    if (i < nwords) p[i] = 0u;
}

// ============================================================================
// K1: masked align-metric  am = score * (iou * mv)^6  via WMMA f32 16x16x4
//     pairwise squared distances (pd-centers and anchor-points vs gt-centers).
// grid = (263, BS), 256 threads = 8 waves; wave w -> anchor tile blockIdx.x*8+w.
// pd_scores is touched once per (b,a): the 32 per-lane label-gathers all hit
// the anchor's own contiguous 320B score row (cache-resident).
// ============================================================================
__global__ __launch_bounds__(256) void k1_metric(
    const float* __restrict__ pd_scores,    // (BS, NA, NC)
    const float* __restrict__ pd_circles,   // (BS, NA, 3)
    const float* __restrict__ anc_points,   // (NA, 2)
    const int*   __restrict__ gt_labels,    // (BS, M)
    const float* __restrict__ gt_circles,   // (BS, M, 3)
    const float* __restrict__ mask_gt,      // (BS, M)
    float* __restrict__ ws_am)              // (BS, M, NA)
{
    __shared__ float sgx[M_GT], sgy[M_GT], sgr[M_GT], sgm[M_GT];
    __shared__ int   sgl[M_GT];
    const int b   = blockIdx.y;
    const int tid = threadIdx.x;
    if (tid < M_GT) {
        sgx[tid] = gt_circles[(b * M_GT + tid) * 3 + 0];
        sgy[tid] = gt_circles[(b * M_GT + tid) * 3 + 1];
        sgr[tid] = gt_circles[(b * M_GT + tid) * 3 + 2];
        sgm[tid] = mask_gt[b * M_GT + tid];
        int gl = gt_labels[b * M_GT + tid];
        sgl[tid] = gl < 0 ? 0 : (gl > NC - 1 ? NC - 1 : gl);
    }
    __syncthreads();

    const int wave = tid >> 5;
    const int lane = tid & 31;
    const int t    = blockIdx.x * 8 + wave;
    if (t >= TILES) return;                 // uniform per wave -> EXEC all-ones

    const int n  = lane & 15;               // tile column / B-matrix N
    const int kh = lane >> 4;               // K-half select for A/B layout
    const int a  = t * 16 + n;

    const float px = pd_circles[((size_t)b * NA + a) * 3 + 0];
    const float py = pd_circles[((size_t)b * NA + a) * 3 + 1];
    const float pr = pd_circles[((size_t)b * NA + a) * 3 + 2];
    const float ax = anc_points[a * 2 + 0];
    const float ay = anc_points[a * 2 + 1];
    const float* srow = pd_scores + ((size_t)b * NA + a) * NC;  // this anchor's scores

    // B col n = (x, y, 1, x^2+y^2); reg r holds K=r (lanes 0-15) / K=r+2 (16-31)
    v2f Bp, Ba;
    Bp.x = kh ? 1.0f : px;
    Bp.y = kh ? (px * px + py * py) : py;
    Ba.x = kh ? 1.0f : ax;
    Ba.y = kh ? (ax * ax + ay * ay) : ay;

    #pragma unroll
    for (int g = 0; g < 4; ++g) {
        // A row m = (-2gx, -2gy, gx^2+gy^2, 1); lanes 0-15: K=0,1; 16-31: K=2,3
        const int   ma = g * 16 + n;
        const float gx = sgx[ma], gy = sgy[ma];
        v2f A;
        A.x = kh ? (gx * gx + gy * gy) : (-2.0f * gx);
        A.y = kh ? 1.0f                : (-2.0f * gy);

        v8f zero = {};
        v8f d2p = __builtin_amdgcn_wmma_f32_16x16x4_f32(
            false, A, false, Bp, (short)0, zero, false, false);   // |g - pd|^2
        v8f d2a = __builtin_amdgcn_wmma_f32_16x16x4_f32(
            false, A, false, Ba, (short)0, zero, false, false);   // |g - anc|^2

        #pragma unroll
        for (int i = 0; i < 8; ++i) {
            const int   m  = g * 16 + i + 8 * kh;   // C/D row for reg i
            const float r0 = sgr[m];
            const float mv = ((d2a[i] < r0 * r0) ? 1.0f : 0.0f) * sgm[m];
            const float o  = circle_iou(d2p[i], r0, pr) * mv;
            const float s  = srow[sgl[m]];
            const float o3 = o * o * o;
            ws_am[((size_t)(b * M_GT + m)) * NA + a] = s * o3 * o3;
        }
    }
}

// ============================================================================
// K2: per-(b,m) row top-13 of align_metric; set bit m for surviving anchors.
// ============================================================================
__global__ __launch_bounds__(256) void k2_topk(
    const float* __restrict__ ws_am,
    const float* __restrict__ anc_points,
    const float* __restrict__ gt_circles,
    const float* __restrict__ mask_gt,
    unsigned int* __restrict__ mask_w)      // (BS*NA) x 2 words (u64 bitmask)
{
    const int row = blockIdx.x;             // b*64 + m
    const int b   = row >> 6;
    const int m   = row & 63;
    if (mask_gt[row] == 0.0f) return;       // counts>1 path -> no positions

    const int tid = threadIdx.x;
    const float gx = gt_circles[row * 3 + 0];
    const float gy = gt_circles[row * 3 + 1];
    const float gr = gt_circles[row * 3 + 2];

    // register-resident sorted (desc) top-13; ties keep earlier (lower) index
    float tv[TOPK]; int ta[TOPK];
    #pragma unroll
    for (int i = 0; i < TOPK; ++i) { tv[i] = -1.0f; ta[i] = 0x7fffffff; }

    const float* amrow = ws_am + (size_t)row * NA;
    for (int a = tid; a < NA; a += 256) {
        const float am = amrow[a];
        if (am > tv[TOPK - 1]) {            // branch-free shift-insert network
            #pragma unroll
            for (int q = TOPK - 1; q > 0; --q) {
                const bool up  = am > tv[q - 1];
                const bool ins = (!up) && (am > tv[q]);
                tv[q] = up ? tv[q - 1] : (ins ? am : tv[q]);
                ta[q] = up ? ta[q - 1] : (ins ? a  : ta[q]);
            }
            if (am > tv[0]) { tv[0] = am; ta[0] = a; }
        }
    }

    __shared__ float cv[256 * TOPK];
    __shared__ int   ci[256 * TOPK];
    __shared__ int   hp[256];
    __shared__ float rv[256];
    __shared__ int   ri[256];
    __shared__ int   rt[256];
    __shared__ int   topa[TOPK];
    #pragma unroll
    for (int i = 0; i < TOPK; ++i) { cv[tid * TOPK + i] = tv[i]; ci[tid * TOPK + i] = ta[i]; }
    hp[tid] = 0;
    __syncthreads();

    for (int r = 0; r < TOPK; ++r) {
        const int p = hp[tid];
        rv[tid] = (p < TOPK) ? cv[tid * TOPK + p] : -2.0f;
        ri[tid] = (p < TOPK) ? ci[tid * TOPK + p] : 0x7fffffff;
        rt[tid] = tid;
        __syncthreads();
        for (int s = 128; s > 0; s >>= 1) {
            if (tid < s) {
                const float v1 = rv[tid + s];
                const int   i1 = ri[tid + s];
                if (v1 > rv[tid] || (v1 == rv[tid] && i1 < ri[tid])) {
                    rv[tid] = v1; ri[tid] = i1; rt[tid] = rt[tid + s];
                }
            }
            __syncthreads();
        }
        if (tid == 0) { topa[r] = ri[0]; hp[rt[0]]++; }
        __syncthreads();
    }

    if (tid < TOPK) {
        const int a = topa[tid];
        const float dx = anc_points[a * 2 + 0] - gx;
        const float dy = anc_points[a * 2 + 1] - gy;
        if (dx * dx + dy * dy < gr * gr) {  // mask_in_gts
            unsigned int* w = mask_w + ((size_t)b * NA + a) * 2;
            if (m < 32) atomicOr(w,     1u << m);
            else        atomicOr(w + 1, 1u << (m - 32));
        }
    }
}

// ============================================================================
// K3: per-anchor conflict resolution + most outputs + pos_align/pos_ov maxes.
// Overlaps recomputed on demand (only for assigned / multi-assigned anchors).
// ============================================================================
__global__ __launch_bounds__(256) void k3_resolve(
    const unsigned int* __restrict__ mask_w,
    const float* __restrict__ ws_am,
    const float* __restrict__ pd_circles,
    const float* __restrict__ anc_points,
    const int*   __restrict__ gt_labels,
    const float* __restrict__ gt_circles,
    const float* __restrict__ mask_gt,
    float* __restrict__ pos_align,
    float* __restrict__ pos_ov,
    float* __restrict__ am_star,
    int*   __restrict__ lab_i32,
    int*   __restrict__ tgt_i32,
    float* __restrict__ out)
{
    const int idx = blockIdx.x * 256 + threadIdx.x;
    if (idx >= BS * NA) return;
    const int b = idx / NA;
    const int a = idx - b * NA;

    const unsigned int lo = mask_w[(size_t)idx * 2];
    const unsigned int hi = mask_w[(size_t)idx * 2 + 1];
    const int fg = __popc(lo) + __popc(hi);

    const float px = pd_circles[(size_t)idx * 3 + 0];
    const float py = pd_circles[(size_t)idx * 3 + 1];
    const float pr = pd_circles[(size_t)idx * 3 + 2];
    const float ax = anc_points[a * 2 + 0];
    const float ay = anc_points[a * 2 + 1];

    int tgt = 0;
    if (fg > 1) {                            // argmax_m of masked overlaps
        float best = -1.0f; int bm = 0;
        for (int mm = 0; mm < M_GT; ++mm) {
            const float cx = gt_circles[(b * M_GT + mm) * 3 + 0];
            const float cy = gt_circles[(b * M_GT + mm) * 3 + 1];
            const float cr = gt_circles[(b * M_GT + mm) * 3 + 2];
            const float dax = ax - cx, day = ay - cy;
            const float mv = ((dax * dax + day * day < cr * cr) ? 1.0f : 0.0f)
                           * mask_gt[b * M_GT + mm];
            const float dx = px - cx, dy = py - cy;
            const float v  = circle_iou(dx * dx + dy * dy, cr, pr) * mv;
            if (v > best) { best = v; bm = mm; }
        }
        tgt = bm;
    } else if (fg == 1) {
        tgt = lo ? (__ffs(lo) - 1) : (__ffs(hi) + 31);
    }

    const int fgf  = (fg > 0) ? 1 : 0;
    const int lraw = gt_labels[b * M_GT + tgt];
    const int tl   = lraw < 0 ? 0 : lraw;               // clip(..., 0, None)

    float amv = 0.0f;
    if (fgf) {
        const float cx = gt_circles[(b * M_GT + tgt) * 3 + 0];
        const float cy = gt_circles[(b * M_GT + tgt) * 3 + 1];
        const float cr = gt_circles[(b * M_GT + tgt) * 3 + 2];
        const float dax = ax - cx, day = ay - cy;
        const float mv = ((dax * dax + day * day < cr * cr) ? 1.0f : 0.0f)
                       * mask_gt[b * M_GT + tgt];
        const float dx = px - cx, dy = py - cy;
        const float ov = circle_iou(dx * dx + dy * dy, cr, pr) * mv;
        amv = ws_am[((size_t)(b * M_GT + tgt)) * NA + a];
        // values >= 0: float bits are int-monotonic
        atomicMax((int*)&pos_align[b * M_GT + tgt], __float_as_int(amv));
        atomicMax((int*)&pos_ov[b * M_GT + tgt],    __float_as_int(ov));
    }
    am_star[idx] = amv;
    lab_i32[idx] = tl;
    tgt_i32[idx] = tgt;

    out[OFF_LAB + idx]         = (float)tl;
    out[OFF_CIR + idx * 3 + 0] = gt_circles[(b * M_GT + tgt) * 3 + 0];
    out[OFF_CIR + idx * 3 + 1] = gt_circles[(b * M_GT + tgt) * 3 + 1];
    out[OFF_CIR + idx * 3 + 2] = gt_circles[(b * M_GT + tgt) * 3 + 2];
    out[OFF_FG  + idx]         = (float)fgf;
    out[OFF_IDX + idx]         = (float)tgt;
}

// ============================================================================
// K4: target_scores = one_hot(label) * am* * pos_ov/(pos_align+eps)
// ============================================================================
__global__ __launch_bounds__(256) void k4_scores(
    const float* __restrict__ am_star,
    const int*   __restrict__ lab_i32,
    const int*   __restrict__ tgt_i32,
    const float* __restrict__ pos_align,
    const float* __restrict__ pos_ov,
    float* __restrict__ out)
{
    const long long e = (long long)blockIdx.x * 256 + threadIdx.x;
    if (e >= (long long)BS * NA * NC) return;
    const int ba = (int)(e / NC);
    const int c  = (int)(e - (long long)ba * NC);
    float v = 0.0f;
    if (c == lab_i32[ba]) {
        const int b   = ba / NA;
        const int tgt = tgt_i32[ba];
        const float r = pos_ov[b * M_GT + tgt] / (pos_align[b * M_GT + tgt] + EPSF);
        v = am_star[ba] * r;                 // 0 when anchor is background
    }
    out[OFF_SCO + e] = v;
}

// ============================================================================
extern "C" void kernel_launch(void* const* d_in, const int* in_sizes, int n_in,
                              void* d_out, int out_size, void* d_ws, size_t ws_size,
                              hipStream_t stream)
{
    const float* pd_scores  = (const float*)d_in[0];
    const float* pd_circles = (const float*)d_in[1];
    const float* anc_points = (const float*)d_in[2];
    const int*   gt_labels  = (const int*)  d_in[3];
    const float* gt_circles = (const float*)d_in[4];
    const float* mask_gt    = (const float*)d_in[5];
    float* out = (float*)d_out;
    char*  ws  = (char*)d_ws;

    float*        ws_am  = (float*)(ws + WS_AMB);
    unsigned int* mask_w = (unsigned int*)(ws + WS_MASK);
    float*        pa     = (float*)(ws + WS_PA);
    float*        po     = (float*)(ws + WS_PO);
    float*        am_st  = (float*)(ws + WS_AMS);
    int*          lab32  = (int*)(ws + WS_LAB);
    int*          tgt32  = (int*)(ws + WS_TGT);

    // K0: zero bitmask + pos_align + pos_ov (contiguous)
    const int nwords = (int)((MASK_BYTES + 2 * POS_BYTES) / 4);
    k0_zero<<<(nwords + 255) / 256, 256, 0, stream>>>((unsigned int*)(ws + WS_MASK), nwords);

    // K1: WMMA distances + branch-free IoU + fused score gather -> align metric
    dim3 g1((TILES + 7) / 8, BS);
    k1_metric<<<g1, 256, 0, stream>>>(pd_scores, pd_circles, anc_points,
                                      gt_labels, gt_circles, mask_gt, ws_am);

    // K2: per-row top-13 -> positive-candidate bitmask (coalesced stream)
    k2_topk<<<BS * M_GT, 256, 0, stream>>>(ws_am, anc_points, gt_circles, mask_gt, mask_w);

    // K3: per-anchor resolution + outputs + pos maxes
    k3_resolve<<<(BS * NA + 255) / 256, 256, 0, stream>>>(
        mask_w, ws_am, pd_circles, anc_points, gt_labels, gt_circles, mask_gt,
        pa, po, am_st, lab32, tgt32, out);

    // K4: normalized target scores
    const long long tot = (long long)BS * NA * NC;
    k4_scores<<<(int)((tot + 255) / 256), 256, 0, stream>>>(am_st, lab32, tgt32, pa, po, out);
}